// LightGCN_28157805592710
// MI455X (gfx1250) — compile-verified
//
#include <hip/hip_runtime.h>

// LightGCN on MI455X (gfx1250, wave32).
// Roofline: ~3.6 GB of gather+atomic traffic, ~0.5 GFLOP -> memory/atomic bound.
// Working set (cur 51MB + tmp 51MB + acc 51MB + edges 7MB) fits in 192MB L2, so
// SpMM atomics and gathers resolve in L2; HBM only sees one input read + output write.
//
// d_ws layout: cur[N*128 f32] | tmp[N*128 f32]   (needs 102.4 MB)
// d_out doubles as the running-sum accumulator (ego + normalized layers), scaled
// by 1/4 in the final normalize pass.

#define N_NODES 100000
#define N_EDGES 600000
#define EMB     128
#define NLAYERS 3
#define EPS_F   1e-12f

typedef __attribute__((ext_vector_type(2))) float v2f;
typedef __attribute__((ext_vector_type(8))) float v8f;

typedef __attribute__((address_space(1))) int gint_t;   // global int
typedef __attribute__((address_space(3))) int lint_t;   // LDS int

#if __has_builtin(__builtin_amdgcn_wmma_f32_16x16x4_f32)
#define USE_WMMA_NORM 1
#endif
#if __has_builtin(__builtin_amdgcn_global_load_async_to_lds_b32)
#define USE_ASYNC_LDS 1
#endif

static __device__ __forceinline__ void wait_async0() {
#if __has_builtin(__builtin_amdgcn_s_wait_asynccnt)
  __builtin_amdgcn_s_wait_asynccnt(0);
#else
  asm volatile("s_wait_asynccnt 0" ::: "memory");
#endif
}

// One 32-bit async global->LDS copy (ASYNCcnt-tracked); sync fallback.
static __device__ __forceinline__ void async_ld_b32(const int* gptr, int* lptr) {
#ifdef USE_ASYNC_LDS
  gint_t* g = const_cast<gint_t*>((const gint_t*)gptr);  // addrspacecast, then drop const
  lint_t* l = (lint_t*)lptr;                             // generic LDS -> AS3
  __builtin_amdgcn_global_load_async_to_lds_b32(g, l, 0, 0);
#else
  *lptr = *gptr;
#endif
}

// acc = ego, cur = nei, tmp = 0. Grid covers exactly N*128/4 float4 elements.
__global__ __launch_bounds__(256) void lgcn_init(const float4* __restrict__ ego,
                                                 const float4* __restrict__ nei,
                                                 float4* __restrict__ acc,
                                                 float4* __restrict__ cur,
                                                 float4* __restrict__ tmp) {
  int i = blockIdx.x * 256 + threadIdx.x;
  acc[i] = ego[i];
  cur[i] = nei[i];
  tmp[i] = make_float4(0.f, 0.f, 0.f, 0.f);
}

// COO SpMM: out[row] += val * x[col]. One wave32 handles one edge-row at a time
// (32 lanes x float4 = 128 floats). Edge lists staged block-wise into LDS via
// async-to-LDS loads (ASYNCcnt), gathers hit L2, scatters via global_atomic_add_f32.
__global__ __launch_bounds__(256) void lgcn_spmm(const int* __restrict__ rows,
                                                 const int* __restrict__ cols,
                                                 const float* __restrict__ vals,
                                                 const float* __restrict__ x,
                                                 float* __restrict__ out,
                                                 int nEdges) {
  __shared__ int   s_rows[256];
  __shared__ int   s_cols[256];
  __shared__ float s_vals[256];
  const int t    = threadIdx.x;
  const int base = blockIdx.x * 256;
  const int e    = base + t;

  if (e < nEdges) {
    async_ld_b32(rows + e, &s_rows[t]);
    async_ld_b32(cols + e, &s_cols[t]);
    async_ld_b32((const int*)(const void*)(vals + e), (int*)&s_vals[t]);
  }
#ifdef USE_ASYNC_LDS
  wait_async0();
#endif
  __syncthreads();

  const int lane = t & 31;
  const int wave = t >> 5;
  int nloc = nEdges - base; if (nloc > 256) nloc = 256;
  int jend = (wave + 1) * 32; if (jend > nloc) jend = nloc;

  for (int j = wave * 32; j < jend; ++j) {
    const int   r = s_rows[j];
    const int   c = s_cols[j];
    const float v = s_vals[j];
    if (j + 1 < jend) {  // global_prefetch of next gather row
      __builtin_prefetch(x + (size_t)s_cols[j + 1] * EMB + lane * 4, 0, 0);
    }
    const float4* xv = (const float4*)(x + (size_t)c * EMB);
    float4 m = xv[lane];
    float* o = out + (size_t)r * EMB + (size_t)lane * 4;
    atomicAdd(o + 0, v * m.x);
    atomicAdd(o + 1, v * m.y);
    atomicAdd(o + 2, v * m.z);
    atomicAdd(o + 3, v * m.w);
  }
}

// Normalize 16 rows per block. Row sum-of-squares computed on the matrix unit:
// D = (X.^2)(16x128) * ones(128x16) via 32 chained V_WMMA_F32_16X16X4_F32 steps
// (every column of D equals the row sum; f32 accumulate, exact precision).
// Then: cur = n, acc += n, tmp = 0; on the last layer: out = (acc + n) / 4.
__global__ __launch_bounds__(256) void lgcn_norm(float* __restrict__ tmp,
                                                 float* __restrict__ cur,
                                                 float* __restrict__ acc,
                                                 int last) {
  __shared__ float tile[16 * 130];  // stride 130 -> conflict-free column reads
  __shared__ float norms[16];
  const int t = threadIdx.x;
  const size_t base4 = (size_t)blockIdx.x * (16 * EMB / 4);  // float4 index

  const float4* gin = (const float4*)tmp;
  float4 a0 = gin[base4 + t * 2 + 0];
  float4 a1 = gin[base4 + t * 2 + 1];
  const int row = t >> 4;          // 16 threads per row
  const int col = (t & 15) * 8;    // 8 consecutive elements, same row
  float* tr = &tile[row * 130 + col];
  tr[0] = a0.x; tr[1] = a0.y; tr[2] = a0.z; tr[3] = a0.w;
  tr[4] = a1.x; tr[5] = a1.y; tr[6] = a1.z; tr[7] = a1.w;
  __syncthreads();

  if (t < 32) {  // wave 0, EXEC all-ones
#ifdef USE_WMMA_NORM
    const int lane = t;
    const int r    = lane & 15;
    const int hi   = lane >> 4;  // A 16x4 f32 layout: lanes 0-15 hold K0,K1; 16-31 hold K2,K3
    v8f c = {0.f, 0.f, 0.f, 0.f, 0.f, 0.f, 0.f, 0.f};
    v2f b; b[0] = 1.0f; b[1] = 1.0f;  // ones matrix: layout-independent
    const float* trow = &tile[r * 130 + hi * 2];
#pragma unroll
    for (int k = 0; k < 32; ++k) {
      float x0 = trow[k * 4 + 0];
      float x1 = trow[k * 4 + 1];
      v2f a; a[0] = x0 * x0; a[1] = x1 * x1;
      c = __builtin_amdgcn_wmma_f32_16x16x4_f32(false, a, false, b,
                                                (short)0, c, false, false);
    }
    // D layout: VGPR i, lanes 0-15 -> row i; lanes 16-31 -> row 8+i. All columns equal.
    if (lane == 0) {
#pragma unroll
      for (int i = 0; i < 8; ++i) norms[i] = c[i];
    } else if (lane == 16) {
#pragma unroll
      for (int i = 0; i < 8; ++i) norms[8 + i] = c[i];
    }
#else
    if (t < 16) {
      float s = 0.f;
      for (int k = 0; k < EMB; ++k) { float v = tile[t * 130 + k]; s += v * v; }
      norms[t] = s;
    }
#endif
  }
  __syncthreads();

  const float nr  = sqrtf(norms[row]);
  const float inv = 1.0f / fmaxf(nr, EPS_F);
  float4 n0 = make_float4(a0.x * inv, a0.y * inv, a0.z * inv, a0.w * inv);
  float4 n1 = make_float4(a1.x * inv, a1.y * inv, a1.z * inv, a1.w * inv);

  float4* acc4 = (float4*)acc;
  const size_t i0 = base4 + t * 2, i1 = i0 + 1;
  float4 A0 = acc4[i0], A1 = acc4[i1];
  if (last) {
    acc4[i0] = make_float4((A0.x + n0.x) * 0.25f, (A0.y + n0.y) * 0.25f,
                           (A0.z + n0.z) * 0.25f, (A0.w + n0.w) * 0.25f);
    acc4[i1] = make_float4((A1.x + n1.x) * 0.25f, (A1.y + n1.y) * 0.25f,
                           (A1.z + n1.z) * 0.25f, (A1.w + n1.w) * 0.25f);
  } else {
    acc4[i0] = make_float4(A0.x + n0.x, A0.y + n0.y, A0.z + n0.z, A0.w + n0.w);
    acc4[i1] = make_float4(A1.x + n1.x, A1.y + n1.y, A1.z + n1.z, A1.w + n1.w);
    float4* cur4 = (float4*)cur;
    cur4[i0] = n0;
    cur4[i1] = n1;
    float4* tmp4 = (float4*)tmp;
    tmp4[i0] = make_float4(0.f, 0.f, 0.f, 0.f);  // fused re-zero for next SpMM
    tmp4[i1] = make_float4(0.f, 0.f, 0.f, 0.f);
  }
}

extern "C" void kernel_launch(void* const* d_in, const int* in_sizes, int n_in,
                              void* d_out, int out_size, void* d_ws, size_t ws_size,
                              hipStream_t stream) {
  const float* ego  = (const float*)d_in[0];
  const float* nei  = (const float*)d_in[1];
  const int*   rows = (const int*)d_in[2];
  const int*   cols = (const int*)d_in[3];
  const float* vals = (const float*)d_in[4];

  float* acc = (float*)d_out;                       // running-sum accumulator
  float* cur = (float*)d_ws;                        // layer input
  float* tmp = cur + (size_t)N_NODES * EMB;         // SpMM atomic target

  const int initBlocks = (N_NODES * EMB / 4) / 256;       // 12500, exact
  const int spmmBlocks = (N_EDGES + 255) / 256;           // 2344
  const int normBlocks = N_NODES / 16;                    // 6250, exact

  lgcn_init<<<initBlocks, 256, 0, stream>>>((const float4*)ego, (const float4*)nei,
                                            (float4*)acc, (float4*)cur, (float4*)tmp);
  for (int l = 0; l < NLAYERS; ++l) {
    lgcn_spmm<<<spmmBlocks, 256, 0, stream>>>(rows, cols, vals, cur, tmp, N_EDGES);
    lgcn_norm<<<normBlocks, 256, 0, stream>>>(tmp, cur, acc, (l == NLAYERS - 1) ? 1 : 0);
  }
}